// MultiHeadLatentAttention_10445360464499
// MI455X (gfx1250) — compile-verified
//
#include <hip/hip_runtime.h>
#include <hip/hip_bf16.h>
#include <math.h>

// Problem constants (B,S,H,DL,T) = (8, 512, 2048, 512, 4095)
constexpr int Bn  = 8;
constexpr int Sn  = 512;
constexpr int Hn  = 2048;
constexpr int DLn = 512;
constexpr int Tn  = 4095;
constexpr int T1n = 4096;
constexpr int HRn = Hn / 2;                 // 1024 rope pairs per row
constexpr float SCALE = 0.022097086912079608f; // 1/sqrt(2048)

// d_out layout: out (8*2048) | key_cache_out (8*4096*2048) | value_cache_out
constexpr size_t OUT_K = (size_t)Bn * Hn;                       // 16384
constexpr size_t OUT_V = OUT_K + (size_t)Bn * T1n * Hn;         // 67125248

// workspace layout (floats)
constexpr int WS_DKV    = 0;        // 8*1024  (k_latent | v_latent)
constexpr int WS_QLAT   = 8192;     // 8*512
constexpr int WS_KUP    = 12288;    // 8*2048
constexpr int WS_VUP    = 28672;    // 8*2048
constexpr int WS_QUP    = 45056;    // 8*2048
constexpr int WS_KROPE  = 61440;    // 8*2048
constexpr int WS_QROPE  = 77824;    // 8*2048
constexpr int WS_SCORES = 94208;    // 8*4096 (softmax in-place)
constexpr int WS_OUTPRE = 126976;   // 8*2048
constexpr int WS_PART   = 143360;   // 8*32*2048 partial sums (deterministic reduce)

typedef __attribute__((ext_vector_type(2))) float v2f;
typedef __attribute__((ext_vector_type(4))) float vf4;
typedef __attribute__((ext_vector_type(8))) float v8f;

__device__ __forceinline__ float wave_reduce_sum(float v) {
#pragma unroll
  for (int off = 16; off > 0; off >>= 1) v += __shfl_xor(v, off, 32);
  return v;
}

// ---------------------------------------------------------------------------
// Skinny GEMM: C(8 x N) = A(8 x K) * B(K x N) via V_WMMA_F32_16X16X4_F32.
// One wave computes one 16-column tile starting at n0. Rows 8..15 of A are
// zero-filled by lane select (no EXEC divergence; WMMA needs EXEC all-ones).
// A-frag: lanes 0-15 carry (K=k,k+1), lanes 16-31 carry (K=k+2,k+3).
// D layout per ISA: VGPR i -> row i (lanes 0-15) / row i+8 (lanes 16-31).
// ---------------------------------------------------------------------------
__device__ __forceinline__ void wmma_skinny_gemm(
    const float* __restrict__ A, int lda,
    const float* __restrict__ B, int ldb,
    float* __restrict__ C, int ldc,
    int K, int n0)
{
  const int lane = threadIdx.x & 31;
  const int kh   = lane >> 4;     // which K-pair this lane supplies
  const int lsub = lane & 15;
  const int m    = lsub;          // A row (valid for m < 8)
  const bool mv  = (m < Bn);
  const int am   = mv ? m : 0;    // clamp address, zero the value
  const int n    = n0 + lsub;     // B/C column
  v8f acc = {0.f, 0.f, 0.f, 0.f, 0.f, 0.f, 0.f, 0.f};
#pragma unroll 4
  for (int k = 0; k < K; k += 4) {
    const int ka = k + 2 * kh;
    float a0 = A[(size_t)am * lda + ka];
    float a1 = A[(size_t)am * lda + ka + 1];
    v2f a; a.x = mv ? a0 : 0.f; a.y = mv ? a1 : 0.f;
    v2f b; b.x = B[(size_t)(ka + 0) * ldb + n];
           b.y = B[(size_t)(ka + 1) * ldb + n];
    acc = __builtin_amdgcn_wmma_f32_16x16x4_f32(
        false, a, false, b, (short)0, acc, false, false);
  }
  if (kh == 0) {                  // rows 0..7 live in lanes 0-15, VGPRs 0..7
#pragma unroll
    for (int i = 0; i < Bn; ++i) C[(size_t)i * ldc + n] = acc[i];
  }
}

// K1: last-token down-projections. 96 waves: 64 tiles of w_dkv, 32 of w_dq.
__global__ void k_proj_down(const float* __restrict__ x,
                            const float* __restrict__ w_dkv,
                            const float* __restrict__ w_dq,
                            float* __restrict__ ws) {
  const int wave = blockIdx.x * (blockDim.x >> 5) + (threadIdx.x >> 5);
  const float* xlast = x + (size_t)(Sn - 1) * Hn;   // row b at stride S*H
  if (wave < 64) {
    wmma_skinny_gemm(xlast, Sn * Hn, w_dkv, 2 * DLn, ws + WS_DKV, 2 * DLn,
                     Hn, wave * 16);
  } else {
    wmma_skinny_gemm(xlast, Sn * Hn, w_dq, DLn, ws + WS_QLAT, DLn,
                     Hn, (wave - 64) * 16);
  }
}

// K2: up-projections k/v/q. 384 waves (3 matrices x 128 tiles).
__global__ void k_proj_up(const float* __restrict__ w_uk,
                          const float* __restrict__ w_uv,
                          const float* __restrict__ w_uq,
                          float* __restrict__ ws) {
  const int wave = blockIdx.x * (blockDim.x >> 5) + (threadIdx.x >> 5);
  const int mat  = wave >> 7;
  const int n0   = (wave & 127) * 16;
  if (mat == 0)
    wmma_skinny_gemm(ws + WS_DKV, 2 * DLn, w_uk, Hn, ws + WS_KUP, Hn, DLn, n0);
  else if (mat == 1)
    wmma_skinny_gemm(ws + WS_DKV + DLn, 2 * DLn, w_uv, Hn, ws + WS_VUP, Hn, DLn, n0);
  else
    wmma_skinny_gemm(ws + WS_QLAT, DLn, w_uq, Hn, ws + WS_QUP, Hn, DLn, n0);
}

// K3: decoupled RoPE on last-token q and k using freqs row 511.
// 16384 threads: idx<8192 -> q pairs, else k pairs.
__global__ void k_rope(const float* __restrict__ freqs, float* __restrict__ ws) {
  const int idx   = blockIdx.x * blockDim.x + threadIdx.x;  // 0..16383
  const int which = idx >> 13;           // 0=q, 1=k
  const int rem   = idx & 8191;
  const int b     = rem >> 10;
  const int i     = rem & (HRn - 1);
  const float f = freqs[(size_t)(Sn - 1) * HRn + i];
  const float c = cosf(f), s = sinf(f);
  const float* src = ws + (which ? WS_KUP : WS_QUP) + b * Hn;
  float* dst       = ws + (which ? WS_KROPE : WS_QROPE) + b * Hn;
  const float x1 = src[2 * i], x2 = src[2 * i + 1];
  dst[2 * i]     = x1 * c - x2 * s;
  dst[2 * i + 1] = x1 * s + x2 * c;
}

// K4: fused K-cache copy + attention scores. 512 blocks (64/batch), 64 rows
// per block, 256 threads. Reads each K row exactly once.
__global__ void k_scores(const float* __restrict__ key_cache,
                         const float* __restrict__ ws,
                         float* __restrict__ key_out,
                         float* __restrict__ scores) {
  __shared__ float red[8];
  const int b    = blockIdx.x >> 6;
  const int t0   = (blockIdx.x & 63) * 64;
  const int tid  = threadIdx.x;
  const int lane = tid & 31, wv = tid >> 5;
  const int h    = tid * 8;
  const float* qp = ws + WS_QROPE + b * Hn + h;
  const vf4 q0 = *(const vf4*)qp;
  const vf4 q1 = *(const vf4*)(qp + 4);
  for (int r = 0; r < 64; ++r) {
    const int t = t0 + r;
    const float* src = (t < Tn)
        ? key_cache + ((size_t)b * Tn + t) * Hn + h
        : ws + WS_KROPE + b * Hn + h;        // appended k_rope last row
    const vf4 k0 = *(const vf4*)src;
    const vf4 k1 = *(const vf4*)(src + 4);
    float* dst = key_out + ((size_t)b * T1n + t) * Hn + h;
    __builtin_nontemporal_store(k0, (vf4*)dst);
    __builtin_nontemporal_store(k1, (vf4*)(dst + 4));
    float p = q0.x * k0.x + q0.y * k0.y + q0.z * k0.z + q0.w * k0.w
            + q1.x * k1.x + q1.y * k1.y + q1.z * k1.z + q1.w * k1.w;
    p = wave_reduce_sum(p);
    if (lane == 0) red[wv] = p;
    __syncthreads();
    if (tid == 0) {
      float s = 0.f;
#pragma unroll
      for (int w = 0; w < 8; ++w) s += red[w];
      scores[(size_t)b * T1n + t] = s * SCALE;
    }
    __syncthreads();
  }
}

// K5: softmax over 4096 scores per batch, in place. 8 blocks x 256 threads.
__global__ void k_softmax(float* __restrict__ scores) {
  __shared__ float red[8];
  const int b = blockIdx.x, tid = threadIdx.x;
  const int lane = tid & 31, wv = tid >> 5;
  float* s = scores + (size_t)b * T1n;
  float v[16];
  float mx = -INFINITY;
#pragma unroll
  for (int i = 0; i < 16; ++i) { v[i] = s[tid + i * 256]; mx = fmaxf(mx, v[i]); }
#pragma unroll
  for (int o = 16; o > 0; o >>= 1) mx = fmaxf(mx, __shfl_xor(mx, o, 32));
  if (lane == 0) red[wv] = mx;
  __syncthreads();
  float m = red[0];
#pragma unroll
  for (int w = 1; w < 8; ++w) m = fmaxf(m, red[w]);
  __syncthreads();
  float sum = 0.f;
#pragma unroll
  for (int i = 0; i < 16; ++i) { v[i] = expf(v[i] - m); sum += v[i]; }
  sum = wave_reduce_sum(sum);
  if (lane == 0) red[wv] = sum;
  __syncthreads();
  float tot = 0.f;
#pragma unroll
  for (int w = 0; w < 8; ++w) tot += red[w];
  const float inv = 1.f / tot;
#pragma unroll
  for (int i = 0; i < 16; ++i) s[tid + i * 256] = v[i] * inv;
}

// K6: fused V-cache copy + probs·V partials. 256 blocks (32 chunks/batch),
// each block owns 128 t-rows x full H; writes its own partial (deterministic).
__global__ void k_values(const float* __restrict__ value_cache,
                         const float* __restrict__ ws_r,
                         float* __restrict__ value_out,
                         float* __restrict__ part,
                         const float* __restrict__ probs) {
  __shared__ float lp[128];
  const int b     = blockIdx.x >> 5;
  const int chunk = blockIdx.x & 31;
  const int t0    = chunk * 128;
  const int tid   = threadIdx.x;
  if (tid < 128) lp[tid] = probs[(size_t)b * T1n + t0 + tid];
  __syncthreads();
  const int h = tid * 8;
  vf4 a0 = {0.f, 0.f, 0.f, 0.f}, a1 = {0.f, 0.f, 0.f, 0.f};
  for (int r = 0; r < 128; ++r) {
    const int t = t0 + r;
    const float* src = (t < Tn)
        ? value_cache + ((size_t)b * Tn + t) * Hn + h
        : ws_r + WS_VUP + b * Hn + h;        // appended v last row
    const vf4 v0 = *(const vf4*)src;
    const vf4 v1 = *(const vf4*)(src + 4);
    float* dst = value_out + ((size_t)b * T1n + t) * Hn + h;
    __builtin_nontemporal_store(v0, (vf4*)dst);
    __builtin_nontemporal_store(v1, (vf4*)(dst + 4));
    const float p = lp[r];
    a0.x += p * v0.x; a0.y += p * v0.y; a0.z += p * v0.z; a0.w += p * v0.w;
    a1.x += p * v1.x; a1.y += p * v1.y; a1.z += p * v1.z; a1.w += p * v1.w;
  }
  float* pp = part + ((size_t)(b * 32 + chunk)) * Hn + h;
  *(vf4*)pp       = a0;
  *(vf4*)(pp + 4) = a1;
}

// K7: reduce 32 partials -> out_pre (8 x 2048). 16384 threads.
__global__ void k_reduce(float* __restrict__ ws) {
  const int idx = blockIdx.x * blockDim.x + threadIdx.x;  // 0..16383
  const int b = idx >> 11, h = idx & (Hn - 1);
  const float* part = ws + WS_PART;
  float s = 0.f;
#pragma unroll
  for (int c = 0; c < 32; ++c) s += part[((size_t)(b * 32 + c)) * Hn + h];
  ws[WS_OUTPRE + idx] = s;
}

// K8: out = out_pre (8x2048) @ w_out (2048x2048), WMMA. 128 waves.
__global__ void k_final(const float* __restrict__ ws,
                        const float* __restrict__ w_out,
                        float* __restrict__ out) {
  const int wave = blockIdx.x * (blockDim.x >> 5) + (threadIdx.x >> 5);
  wmma_skinny_gemm(ws + WS_OUTPRE, Hn, w_out, Hn, out, Hn, Hn, wave * 16);
}

extern "C" void kernel_launch(void* const* d_in, const int* in_sizes, int n_in,
                              void* d_out, int out_size, void* d_ws, size_t ws_size,
                              hipStream_t stream) {
  const float* x           = (const float*)d_in[0];
  const float* freqs       = (const float*)d_in[1];
  const float* key_cache   = (const float*)d_in[2];
  const float* value_cache = (const float*)d_in[3];
  const float* w_dkv       = (const float*)d_in[4];
  const float* w_dq        = (const float*)d_in[5];
  const float* w_uk        = (const float*)d_in[6];
  const float* w_uv        = (const float*)d_in[7];
  const float* w_uq        = (const float*)d_in[8];
  const float* w_out       = (const float*)d_in[9];
  float* out = (float*)d_out;
  float* ws  = (float*)d_ws;

  k_proj_down<<<24, 128, 0, stream>>>(x, w_dkv, w_dq, ws);
  k_proj_up  <<<96, 128, 0, stream>>>(w_uk, w_uv, w_uq, ws);
  k_rope     <<<64, 256, 0, stream>>>(freqs, ws);
  k_scores   <<<512, 256, 0, stream>>>(key_cache, ws, out + OUT_K, ws + WS_SCORES);
  k_softmax  <<<8, 256, 0, stream>>>(ws + WS_SCORES);
  k_values   <<<256, 256, 0, stream>>>(value_cache, ws, out + OUT_V,
                                       ws + WS_PART, ws + WS_SCORES);
  k_reduce   <<<64, 256, 0, stream>>>(ws);
  k_final    <<<32, 128, 0, stream>>>(ws, w_out, out);
}